// QConv2dSlow_56856777065115
// MI455X (gfx1250) — compile-verified
//
#include <hip/hip_runtime.h>

// ---------------- problem constants ----------------
constexpr int IN_C   = 3;
constexpr int OUT_C  = 8;
constexpr int QDEPTH = 2;
constexpr int WIRES  = 4;
constexpr int DIM    = 16;     // 2^WIRES
constexpr int NB     = 4;      // batch
constexpr int H      = 128;
constexpr int W      = 128;

typedef float v2f __attribute__((ext_vector_type(2)));
typedef float v8f __attribute__((ext_vector_type(8)));

// =====================================================================
// Kernel 1: build the three 16x16 complex entangling unitaries from
// `weights` (depends only on weights -> do it once per launch).
// Thread (ic,col) propagates basis vector e_col through the circuit;
// the result is column `col` of U_ic.
// =====================================================================
__global__ __launch_bounds__(64)
void build_unitaries(const float* __restrict__ w, float* __restrict__ U)
{
    int tid = threadIdx.x;
    if (tid >= IN_C * DIM) return;
    int ic  = tid >> 4;
    int col = tid & 15;

    float cr[DIM], ci[DIM];
    for (int i = 0; i < DIM; ++i) { cr[i] = 0.f; ci[i] = 0.f; }
    cr[col] = 1.f;

    for (int l = 0; l < QDEPTH; ++l) {
        // Rot(phi, theta, omega) on each wire (wire 0 = MSB)
        for (int q = 0; q < WIRES; ++q) {
            const float* wp = w + ((ic * QDEPTH + l) * WIRES + q) * 3;
            float phi = wp[0], th = wp[1], om = wp[2];
            float ct  = cosf(0.5f * th),        st  = sinf(0.5f * th);
            float cpo = cosf(0.5f * (phi + om)), spo = sinf(0.5f * (phi + om));
            float cmo = cosf(0.5f * (phi - om)), smo = sinf(0.5f * (phi - om));
            float m00r =  cpo * ct, m00i = -spo * ct;
            float m01r = -cmo * st, m01i = -smo * st;
            float m10r =  cmo * st, m10i = -smo * st;
            float m11r =  cpo * ct, m11i =  spo * ct;
            int stride = 1 << (3 - q);
            for (int i0 = 0; i0 < DIM; ++i0) {
                if (i0 & stride) continue;
                int i1 = i0 | stride;
                float ar = cr[i0], ai = ci[i0];
                float br = cr[i1], bi = ci[i1];
                cr[i0] = m00r*ar - m00i*ai + m01r*br - m01i*bi;
                ci[i0] = m00r*ai + m00i*ar + m01r*bi + m01i*br;
                cr[i1] = m10r*ar - m10i*ai + m11r*br - m11i*bi;
                ci[i1] = m10r*ai + m10i*ar + m11r*bi + m11i*br;
            }
        }
        // CNOT ring with range r = l % (WIRES-1) + 1
        int r = l % (WIRES - 1) + 1;
        for (int q = 0; q < WIRES; ++q) {
            int cbit = 1 << (3 - q);
            int tbit = 1 << (3 - ((q + r) % WIRES));
            float tr[DIM], ti[DIM];
            for (int i = 0; i < DIM; ++i) { tr[i] = cr[i]; ti[i] = ci[i]; }
            for (int i = 0; i < DIM; ++i) {
                int src = (i & cbit) ? (i ^ tbit) : i;
                cr[i] = tr[src]; ci[i] = ti[src];
            }
        }
    }
    for (int row = 0; row < DIM; ++row) {
        U[((ic * 2 + 0) * DIM + row) * DIM + col] = cr[row];   // Re
        U[((ic * 2 + 1) * DIM + row) * DIM + col] = ci[row];   // Im
    }
}

// =====================================================================
// Kernel 2: one wave (32 lanes) per block, one pixel per lane.
// Per channel: angle computation + RY cascade per-lane (registers),
// then the entangling 16x16 complex mat-vec batched over 16 pixels
// via V_WMMA_F32_16X16X4_F32 (two halves of the wave's 32 pixels).
// =====================================================================
__global__ __launch_bounds__(32)
void qconv_main(const float* __restrict__ x,
                const float* __restrict__ U,
                float* __restrict__ out)
{
    __shared__ float lds_sr[32][DIM + 1];
    __shared__ float lds_si[32][DIM + 1];

    const int lane = threadIdx.x;                 // 0..31
    const int p    = blockIdx.x * 32 + lane;      // pixel id
    const int b    = p >> 14;                     // / (128*128)
    const int rem  = p & 16383;
    const int yy   = rem >> 7;
    const int xx   = rem & 127;

    const int M  = lane & 15;                     // A-row / B-column index
    const int hh = lane >> 4;                     // lane half (K split)

    // per-pixel state |0...0>
    float sr[DIM], si[DIM];
    #pragma unroll
    for (int m = 0; m < DIM; ++m) { sr[m] = 0.f; si[m] = 0.f; }
    sr[0] = 1.f;

    #pragma unroll
    for (int ic = 0; ic < IN_C; ++ic) {
        // ---- squared patch amplitudes (norm cancels in angle ratios) ----
        float pr[DIM];
        const float* xc = x + (b * IN_C + ic) * (H * W);
        #pragma unroll
        for (int t = 0; t < 9; ++t) {
            int di = t / 3, dj = t % 3;
            int iy = yy + di - 1, ix = xx + dj - 1;
            float v = 0.f;
            if (iy >= 0 && iy < H && ix >= 0 && ix < W) v = xc[iy * W + ix];
            v += 0.01f;
            pr[t] = v * v;
        }
        #pragma unroll
        for (int t = 9; t < DIM; ++t) pr[t] = 0.01f * 0.01f;

        // ---- partial-sum tree; sin = sqrt(ratio), cos = sqrt(1-ratio) ----
        float s2[8], s4[4], s8[2], s16;
        #pragma unroll
        for (int t = 0; t < 8; ++t) s2[t] = pr[2*t] + pr[2*t+1];
        #pragma unroll
        for (int t = 0; t < 4; ++t) s4[t] = s2[2*t] + s2[2*t+1];
        #pragma unroll
        for (int t = 0; t < 2; ++t) s8[t] = s4[2*t] + s4[2*t+1];
        s16 = s8[0] + s8[1];

        float cc[15], ss[15];
        {
            float ratio = fminf(fmaxf(s8[1] / fmaxf(s16, 1e-12f), 0.f), 1.f);
            ss[0] = sqrtf(ratio); cc[0] = sqrtf(1.f - ratio);
        }
        #pragma unroll
        for (int t = 0; t < 2; ++t) {
            float ratio = fminf(fmaxf(s4[2*t+1] / fmaxf(s8[t], 1e-12f), 0.f), 1.f);
            ss[1+t] = sqrtf(ratio); cc[1+t] = sqrtf(1.f - ratio);
        }
        #pragma unroll
        for (int t = 0; t < 4; ++t) {
            float ratio = fminf(fmaxf(s2[2*t+1] / fmaxf(s4[t], 1e-12f), 0.f), 1.f);
            ss[3+t] = sqrtf(ratio); cc[3+t] = sqrtf(1.f - ratio);
        }
        #pragma unroll
        for (int t = 0; t < 8; ++t) {
            float ratio = fminf(fmaxf(pr[2*t+1] / fmaxf(s2[t], 1e-12f), 0.f), 1.f);
            ss[7+t] = sqrtf(ratio); cc[7+t] = sqrtf(1.f - ratio);
        }

        // ---- RY cascade (real coefficients: identical on Re and Im) ----
        #pragma unroll
        for (int q = 0; q < WIRES; ++q) {
            int stride = 1 << (3 - q);
            int off    = (1 << q) - 1;
            #pragma unroll
            for (int i0 = 0; i0 < DIM; ++i0) {
                if (i0 & stride) continue;
                int i1 = i0 | stride;
                int t  = i0 >> (4 - q);          // top q bits -> angle index
                float c = cc[off + t], s = ss[off + t];
                float a0 = sr[i0], a1 = sr[i1];
                sr[i0] = c*a0 - s*a1;  sr[i1] = s*a0 + c*a1;
                float b0 = si[i0], b1 = si[i1];
                si[i0] = c*b0 - s*b1;  si[i1] = s*b0 + c*b1;
            }
        }

        // ---- entangling unitary: S_new = U_ic * S via WMMA f32 16x16x4 ----
        // Stage per-lane states into LDS ([pixel][m]).
        #pragma unroll
        for (int m = 0; m < DIM; ++m) { lds_sr[lane][m] = sr[m]; lds_si[lane][m] = si[m]; }
        __syncthreads();

        // A-operand tiles (16x4 f32 layout: lanes 0-15 -> K={k0,k0+1},
        // lanes 16-31 -> K={k0+2,k0+3}); pre-negated Im tiles for the
        // real-part accumulation since f32 WMMA NEG is C-only.
        v2f ar[4], ai[4], ain[4];
        const float* Urow_r = U + ((ic * 2 + 0) * DIM + M) * DIM;
        const float* Urow_i = U + ((ic * 2 + 1) * DIM + M) * DIM;
        #pragma unroll
        for (int t = 0; t < 4; ++t) {
            int k = 4 * t + 2 * hh;
            ar[t]  = (v2f){ Urow_r[k],  Urow_r[k+1] };
            ai[t]  = (v2f){ Urow_i[k],  Urow_i[k+1] };
            ain[t] = (v2f){ -Urow_i[k], -Urow_i[k+1] };
        }

        #pragma unroll
        for (int hf = 0; hf < 2; ++hf) {
            int col = hf * 16 + M;               // pixel forming column M
            v2f br[4], bi[4];
            #pragma unroll
            for (int t = 0; t < 4; ++t) {
                int k = 4 * t + 2 * hh;          // B 4x16: K split by lane half
                br[t] = (v2f){ lds_sr[col][k], lds_sr[col][k+1] };
                bi[t] = (v2f){ lds_si[col][k], lds_si[col][k+1] };
            }
            v8f accr = {};
            v8f acci = {};
            #pragma unroll
            for (int t = 0; t < 4; ++t) {
                accr = __builtin_amdgcn_wmma_f32_16x16x4_f32(false, ar[t],  false, br[t], (short)0, accr, false, false);
                acci = __builtin_amdgcn_wmma_f32_16x16x4_f32(false, ar[t],  false, bi[t], (short)0, acci, false, false);
            }
            #pragma unroll
            for (int t = 0; t < 4; ++t) {
                accr = __builtin_amdgcn_wmma_f32_16x16x4_f32(false, ain[t], false, bi[t], (short)0, accr, false, false);
                acci = __builtin_amdgcn_wmma_f32_16x16x4_f32(false, ai[t],  false, br[t], (short)0, acci, false, false);
            }
            // D (C-layout): lane holds rows M = v + 8*hh of column (lane&15)
            #pragma unroll
            for (int v = 0; v < 8; ++v) {
                lds_sr[hf * 16 + M][v + 8 * hh] = accr[v];
                lds_si[hf * 16 + M][v + 8 * hh] = acci[v];
            }
        }
        __syncthreads();
        #pragma unroll
        for (int m = 0; m < DIM; ++m) { sr[m] = lds_sr[lane][m]; si[m] = lds_si[lane][m]; }
        __syncthreads();
    }

    // ---- probabilities -> clipped output (first 8 basis states) ----
    float* ob = out + b * (OUT_C * H * W) + yy * W + xx;
    #pragma unroll
    for (int oc = 0; oc < OUT_C; ++oc) {
        float pv = sr[oc] * sr[oc] + si[oc] * si[oc];
        pv = fminf(fmaxf(pv * (DIM * 0.5f), 0.f), 1.f);
        ob[oc * (H * W)] = pv;
    }
}

// =====================================================================
extern "C" void kernel_launch(void* const* d_in, const int* in_sizes, int n_in,
                              void* d_out, int out_size, void* d_ws, size_t ws_size,
                              hipStream_t stream)
{
    const float* x = (const float*)d_in[0];   // (4,3,128,128) f32
    const float* w = (const float*)d_in[1];   // (3,2,4,3)     f32
    float* out = (float*)d_out;               // (4,8,128,128) f32
    float* U   = (float*)d_ws;                // 3*2*16*16 floats = 6 KB scratch

    build_unitaries<<<1, 64, 0, stream>>>(w, U);

    const int npix = NB * H * W;              // 65536 pixels
    qconv_main<<<npix / 32, 32, 0, stream>>>(x, U, out);
}